// VarianceAdaptor_42691974922416
// MI455X (gfx1250) — compile-verified
//
#include <hip/hip_runtime.h>

typedef __attribute__((ext_vector_type(2))) float v2f;
typedef __attribute__((ext_vector_type(8))) float v8f;

#define F_DIM 256
#define KW 3
#define KDIM (KW * F_DIM) /* 768 */

// Fused conv1d(K=3, SAME) + bias + ReLU + LayerNorm over F=256.
// Workgroup = 512 threads (16 waves); computes a 16-row x 256-col block.
// Wave w computes the 16x16 tile for columns [16w,16w+16) via fp32 WMMA.
__launch_bounds__(512)
__global__ void conv_ln_kernel(const float* __restrict__ x,     // (B*Tlen, 256)
                               const float* __restrict__ W,     // (768, 256) = w flattened (K,H,F)
                               const float* __restrict__ bias,  // (256)
                               const float* __restrict__ g,     // ln gamma
                               const float* __restrict__ beta,  // ln beta
                               float* __restrict__ y,           // (B*Tlen, 256)
                               int Tlen)
{
    __shared__ float lds[16 * KDIM];        // A panel; reused as 16x256 output tile
    __shared__ float ldsMu[16];
    __shared__ float ldsRstd[16];

    const int tid  = threadIdx.x;
    const int m0   = blockIdx.x * 16;       // Tlen multiple of 16 -> no batch crossing
    const int bIdx = m0 / Tlen;
    const int t0   = m0 % Tlen;
    const float* xb = x + (size_t)bIdx * Tlen * F_DIM;

    // Stage padded A panel: A[r][k], k = kseg*256 + c, value x[b, t0+r+kseg-1, c]
    for (int e = tid; e < 16 * KDIM; e += 512) {
        int r    = e / KDIM;
        int k    = e - r * KDIM;
        int kseg = k >> 8;                  // 0..2 (conv tap)
        int c    = k & 255;                 // channel
        int ts   = t0 + r + kseg - 1;       // SAME padding
        float v  = 0.0f;
        if (ts >= 0 && ts < Tlen) v = xb[(size_t)ts * F_DIM + c];
        lds[e] = v;
    }
    __syncthreads();

    const int wave  = tid >> 5;
    const int lane  = tid & 31;
    const int n     = lane & 15;            // row (A) / col-in-tile (B,C)
    const int col   = wave * 16 + n;        // global output column
    const int khalf = (lane >> 4) << 1;     // lanes 0-15 -> K{0,1}; 16-31 -> K{2,3}

    v8f acc = {};
    const float* ldsArow = lds + n * KDIM;
    #pragma unroll 4
    for (int kk = 0; kk < KDIM; kk += 4) {
        v2f a, bm;
        a.x  = ldsArow[kk + khalf];
        a.y  = ldsArow[kk + khalf + 1];
        bm.x = W[(size_t)(kk + khalf) * F_DIM + col];
        bm.y = W[(size_t)(kk + khalf + 1) * F_DIM + col];
        acc = __builtin_amdgcn_wmma_f32_16x16x4_f32(false, a, false, bm,
                                                    (short)0, acc, false, false);
    }
    __syncthreads();    // everyone done reading the A panel

    // bias + ReLU, scatter C tile into LDS (C layout: vgpr j -> M=j (lanes<16) / M=8+j)
    {
        const float bc   = bias[col];
        const int mbase  = (lane >> 4) * 8;
        #pragma unroll
        for (int j = 0; j < 8; ++j) {
            float v = acc[j] + bc;
            v = fmaxf(v, 0.0f);
            lds[(mbase + j) * F_DIM + col] = v;
        }
    }
    __syncthreads();

    // LayerNorm stats: wave w reduces row w (256 elems, 8 per lane, shuffle tree)
    {
        const float* row = lds + wave * F_DIM;
        float s = 0.0f;
        #pragma unroll
        for (int j = 0; j < 8; ++j) s += row[lane + 32 * j];
        #pragma unroll
        for (int off = 16; off > 0; off >>= 1) s += __shfl_xor(s, off, 32);
        const float mu = s * (1.0f / F_DIM);
        float s2 = 0.0f;
        #pragma unroll
        for (int j = 0; j < 8; ++j) {
            float d = row[lane + 32 * j] - mu;
            s2 += d * d;
        }
        #pragma unroll
        for (int off = 16; off > 0; off >>= 1) s2 += __shfl_xor(s2, off, 32);
        if (lane == 0) {
            ldsMu[wave]   = mu;
            ldsRstd[wave] = rsqrtf(s2 * (1.0f / F_DIM) + 1e-5f);
        }
    }
    __syncthreads();

    // Coalesced LN-affine writeout
    for (int e = tid; e < 16 * F_DIM; e += 512) {
        int r = e >> 8;
        int f = e & 255;
        float v = (lds[e] - ldsMu[r]) * ldsRstd[r] * g[f] + beta[f];
        y[(size_t)(m0 + r) * F_DIM + f] = v;
    }
}

// Per-row 256-dot with lw, + lb. Duration mode: clip(round(exp(v)-1),0) + int cast.
__launch_bounds__(256)
__global__ void linear_head_kernel(const float* __restrict__ h,
                                   const float* __restrict__ lw,
                                   const float* __restrict__ lb,
                                   float* __restrict__ out,
                                   int* __restrict__ dur_int,   // null => plain linear
                                   int M)
{
    const int wave = threadIdx.x >> 5;
    const int lane = threadIdx.x & 31;
    const int m = blockIdx.x * 8 + wave;
    if (m >= M) return;
    const float* row = h + (size_t)m * F_DIM;
    float s = 0.0f;
    #pragma unroll
    for (int j = 0; j < 8; ++j) {
        int f = lane + 32 * j;
        s += row[f] * lw[f];
    }
    #pragma unroll
    for (int off = 16; off > 0; off >>= 1) s += __shfl_xor(s, off, 32);
    if (lane == 0) {
        float v = s + lb[0];
        if (dur_int) {
            float pd = fmaxf(rintf(expf(v) - 1.0f), 0.0f);  // rint = half-even (jnp.round)
            out[m] = pd;
            dur_int[m] = (int)pd;
        } else {
            out[m] = v;
        }
    }
}

// In-place inclusive cumsum of durations per batch (32 batches x 512).
__global__ void cumsum_kernel(int* __restrict__ dur)
{
    int b = threadIdx.x;
    if (b >= 32) return;
    int* p = dur + b * 512;
    int s = 0;
    for (int t = 0; t < 512; ++t) { s += p[t]; p[t] = s; }
}

// length_regulate: tok = searchsorted(ends, l, 'right'), clamp, gather, mask.
__launch_bounds__(256)
__global__ void length_regulate_kernel(const float* __restrict__ enc,  // (B,512,256)
                                       const int* __restrict__ ends,   // (B,512)
                                       float* __restrict__ lr)         // (B,2048,256)
{
    const int wave = threadIdx.x >> 5;
    const int lane = threadIdx.x & 31;
    const int idx  = blockIdx.x * 8 + wave;   // 0..B*L-1
    const int b    = idx >> 11;
    const int l    = idx & 2047;
    const int* e = ends + b * 512;
    int lo = 0, hi = 512;
    while (lo < hi) {
        int mid = (lo + hi) >> 1;
        if (e[mid] <= l) lo = mid + 1; else hi = mid;
    }
    const int tok    = lo < 511 ? lo : 511;
    const float mask = (l < e[511]) ? 1.0f : 0.0f;
    const float* src = enc + ((size_t)b * 512 + tok) * F_DIM;
    float* dst       = lr + (size_t)idx * F_DIM;
    #pragma unroll
    for (int j = 0; j < 8; ++j) {
        int f = lane + 32 * j;
        dst[f] = mask * src[f];
    }
}

// out = lr_out + pred_pitch + pred_energy (broadcast scalars per row)
__global__ void combine_kernel(const float* __restrict__ lr,
                               const float* __restrict__ pp,
                               const float* __restrict__ pe,
                               float* __restrict__ out)
{
    size_t i = (size_t)blockIdx.x * blockDim.x + threadIdx.x;
    size_t row = i >> 8;
    out[i] = lr[i] + pp[row] + pe[row];
}

struct PredParams {
    const float *w1, *b1, *w2, *b2, *g, *be, *lw, *lb;
};

static inline PredParams get_params(void* const* d_in, int base) {
    PredParams p;
    p.w1 = (const float*)d_in[base + 0];
    p.b1 = (const float*)d_in[base + 1];
    p.w2 = (const float*)d_in[base + 2];
    p.b2 = (const float*)d_in[base + 3];
    p.g  = (const float*)d_in[base + 4];
    p.be = (const float*)d_in[base + 5];
    p.lw = (const float*)d_in[base + 6];
    p.lb = (const float*)d_in[base + 7];
    return p;
}

extern "C" void kernel_launch(void* const* d_in, const int* in_sizes, int n_in,
                              void* d_out, int out_size, void* d_ws, size_t ws_size,
                              hipStream_t stream)
{
    (void)in_sizes; (void)n_in; (void)out_size; (void)ws_size;

    const int B = 32, T = 512, L = 2048;
    const int MT = B * T;    // 16384 rows (duration path)
    const int ML = B * L;    // 65536 rows (pitch/energy path)

    const float* enc = (const float*)d_in[0];
    PredParams dP = get_params(d_in, 1);   // dur
    PredParams pP = get_params(d_in, 9);   // pitch
    PredParams eP = get_params(d_in, 17);  // energy

    // Workspace carve-out (bytes): lr_out | h1 | h2 | dur_int
    char* ws = (char*)d_ws;
    float* lr   = (float*)(ws);                              // 64 MiB
    float* h1   = (float*)(ws + (size_t)ML * F_DIM * 4);     // 64 MiB
    float* h2   = (float*)(ws + (size_t)2 * ML * F_DIM * 4); // 64 MiB
    int*   durI = (int*)  (ws + (size_t)3 * ML * F_DIM * 4); // 64 KiB

    // Output slices (tuple concatenated flat, f32)
    float* out_dur    = (float*)d_out;          // (B,T,1)
    float* out_pitch  = out_dur + MT;           // (B,L,1)
    float* out_energy = out_pitch + ML;         // (B,L,1)
    float* out_main   = out_energy + ML;        // (B,L,256)

    // --- duration predictor on encoder_out ---
    conv_ln_kernel<<<MT / 16, 512, 0, stream>>>(enc, dP.w1, dP.b1, dP.g, dP.be, h1, T);
    conv_ln_kernel<<<MT / 16, 512, 0, stream>>>(h1,  dP.w2, dP.b2, dP.g, dP.be, h2, T);
    linear_head_kernel<<<MT / 8, 256, 0, stream>>>(h2, dP.lw, dP.lb, out_dur, durI, MT);

    // --- length regulate ---
    cumsum_kernel<<<1, 32, 0, stream>>>(durI);
    length_regulate_kernel<<<ML / 8, 256, 0, stream>>>(enc, durI, lr);

    // --- pitch predictor on lr_out ---
    conv_ln_kernel<<<ML / 16, 512, 0, stream>>>(lr, pP.w1, pP.b1, pP.g, pP.be, h1, L);
    conv_ln_kernel<<<ML / 16, 512, 0, stream>>>(h1, pP.w2, pP.b2, pP.g, pP.be, h2, L);
    linear_head_kernel<<<ML / 8, 256, 0, stream>>>(h2, pP.lw, pP.lb, out_pitch, nullptr, ML);

    // --- energy predictor on lr_out ---
    conv_ln_kernel<<<ML / 16, 512, 0, stream>>>(lr, eP.w1, eP.b1, eP.g, eP.be, h1, L);
    conv_ln_kernel<<<ML / 16, 512, 0, stream>>>(h1, eP.w2, eP.b2, eP.g, eP.be, h2, L);
    linear_head_kernel<<<ML / 8, 256, 0, stream>>>(h2, eP.lw, eP.lb, out_energy, nullptr, ML);

    // --- out = lr_out + pitch + energy ---
    combine_kernel<<<(size_t)ML * F_DIM / 256, 256, 0, stream>>>(lr, out_pitch, out_energy, out_main);
}